// Decoder_61821759259085
// MI455X (gfx1250) — compile-verified
//
#include <hip/hip_runtime.h>
#include <stdint.h>

// ---------------------------------------------------------------------------
// SSD Decoder on gfx1250 (MI455X):
//   k1: box decode + softmax  (memory-bound elementwise, ~80MB in / 76MB out)
//   k2: per-(b,c) exact top-K: whole 96KB score row DMA'd into LDS ONCE via
//       CDNA5 async engine (global_load_async_to_lds_b128, depth-4 pipeline,
//       s_wait_asynccnt), hist pass fused with DMA; refine+compact passes are
//       pure LDS; 2-level radix select + bitonic sort on (~key,idx) composite.
//   k3: greedy NMS sequential scan (lax.scan semantics) + flat outputs.
// ---------------------------------------------------------------------------

#define TPB   256
#define TILE  1024          // floats per DMA tile (= TPB lanes * 4 floats)
#define DEPTH 4             // async pipeline depth (tiles in flight)
#define NBIN  4096
#define CANDN 1024
#define KMAX  256

__device__ __forceinline__ uint32_t lds_off32(const void* p) {
  // generic pointer into LDS: low 32 bits == wave-relative LDS byte offset
  return (uint32_t)(uintptr_t)p;
}

__device__ __forceinline__ void async_load_b128(uint32_t lds_byte_off, const void* gptr) {
  uint64_t g = (uint64_t)(uintptr_t)gptr;
  asm volatile("global_load_async_to_lds_b128 %0, %1, off"
               :: "v"(lds_byte_off), "v"(g) : "memory");
}
// s_wait_asynccnt takes an immediate only; async loads complete in order.
__device__ __forceinline__ void wait_async(int rem) {
  switch (rem) {
    case 0:  asm volatile("s_wait_asynccnt 0x0" ::: "memory"); break;
    case 1:  asm volatile("s_wait_asynccnt 0x1" ::: "memory"); break;
    case 2:  asm volatile("s_wait_asynccnt 0x2" ::: "memory"); break;
    case 3:  asm volatile("s_wait_asynccnt 0x3" ::: "memory"); break;
    default: asm volatile("s_wait_asynccnt 0x4" ::: "memory"); break;
  }
}

// ------------------------------ kernel 1 -----------------------------------
template <int CT>
__global__ void __launch_bounds__(TPB)
decode_softmax_kernel(const float* __restrict__ bb_in,   // [B,4,N]
                      const float* __restrict__ sc_in,   // [B,CT,N]
                      const float* __restrict__ dbox,    // [N,4]
                      const float* __restrict__ sxy_p,
                      const float* __restrict__ swh_p,
                      float* __restrict__ boxes,         // [B,N,4] ltrb
                      float* __restrict__ probs,         // [B,CT-1,N]
                      int B, int N) {
  int tid = blockIdx.x * blockDim.x + threadIdx.x;
  if (tid >= B * N) return;
  int b = tid / N, n = tid - b * N;
  float sxy = *sxy_p, swh = *swh_p;

  const float* dp = dbox + (size_t)n * 4;
  float dx = dp[0], dy = dp[1], dw = dp[2], dh = dp[3];
  const float* bp = bb_in + (size_t)b * 4 * N + n;
  float ox = bp[0], oy = bp[(size_t)N], ow = bp[2 * (size_t)N], oh = bp[3 * (size_t)N];

  float cx = sxy * ox * dw + dx;
  float cy = sxy * oy * dh + dy;
  float w  = __expf(swh * ow) * dw;
  float h  = __expf(swh * oh) * dh;
  float4 box = make_float4(cx - 0.5f * w, cy - 0.5f * h, cx + 0.5f * w, cy + 0.5f * h);
  *(float4*)(boxes + (size_t)tid * 4) = box;

  const float* sp = sc_in + (size_t)b * CT * N + n;
  float sv[CT];
  float mx = -INFINITY;
#pragma unroll
  for (int c = 0; c < CT; ++c) { sv[c] = sp[(size_t)c * N]; mx = fmaxf(mx, sv[c]); }
  float sum = 0.f;
#pragma unroll
  for (int c = 0; c < CT; ++c) { sv[c] = __expf(sv[c] - mx); sum += sv[c]; }
  float inv = 1.f / sum;
  float* pp = probs + (size_t)b * (CT - 1) * N + n;
#pragma unroll
  for (int c = 1; c < CT; ++c) pp[(size_t)(c - 1) * N] = sv[c] * inv;
}

__global__ void __launch_bounds__(TPB)
decode_softmax_dyn(const float* __restrict__ bb_in, const float* __restrict__ sc_in,
                   const float* __restrict__ dbox, const float* __restrict__ sxy_p,
                   const float* __restrict__ swh_p, float* __restrict__ boxes,
                   float* __restrict__ probs, int B, int N, int C) {
  int tid = blockIdx.x * blockDim.x + threadIdx.x;
  if (tid >= B * N) return;
  int b = tid / N, n = tid - b * N;
  float sxy = *sxy_p, swh = *swh_p;
  const float* dp = dbox + (size_t)n * 4;
  float dx = dp[0], dy = dp[1], dw = dp[2], dh = dp[3];
  const float* bp = bb_in + (size_t)b * 4 * N + n;
  float cx = sxy * bp[0] * dw + dx;
  float cy = sxy * bp[(size_t)N] * dh + dy;
  float w  = __expf(swh * bp[2 * (size_t)N]) * dw;
  float h  = __expf(swh * bp[3 * (size_t)N]) * dh;
  float4 box = make_float4(cx - 0.5f * w, cy - 0.5f * h, cx + 0.5f * w, cy + 0.5f * h);
  *(float4*)(boxes + (size_t)tid * 4) = box;

  const float* sp = sc_in + (size_t)b * C * N + n;
  float mx = -INFINITY;
  for (int c = 0; c < C; ++c) mx = fmaxf(mx, sp[(size_t)c * N]);
  float sum = 0.f;
  for (int c = 0; c < C; ++c) sum += __expf(sp[(size_t)c * N] - mx);
  float inv = 1.f / sum;
  float* pp = probs + (size_t)b * (C - 1) * N + n;
  for (int c = 1; c < C; ++c) pp[(size_t)(c - 1) * N] = __expf(sp[(size_t)c * N] - mx) * inv;
}

// ------------------------------ kernel 2 -----------------------------------
// one block per (b, class); exact top-K (desc score, asc idx) of probs row.
// Dynamic LDS layout: hist[NBIN] | cand[CANDN] | row[nTiles*TILE]
__global__ void __launch_bounds__(TPB)
topk_kernel(const float* __restrict__ probs,   // [B*CM1, N]
            const float* __restrict__ conf_th, // [CM1]
            float* __restrict__ tk_sc,         // [B*CM1, K]
            int* __restrict__ tk_id,           // [B*CM1, K]
            int CM1, int N, int K) {
  extern __shared__ uint8_t smem[];
  uint32_t* hist = (uint32_t*)smem;                                   // 16 KB
  unsigned long long* cand = (unsigned long long*)(smem + NBIN * 4);  //  8 KB
  float* row = (float*)(smem + NBIN * 4 + CANDN * 8);                 // padded row
  __shared__ uint32_t s_cnt, s_b1, s_cAbove, s_thresh, s_flag;

  const int blk = blockIdx.x;
  const int c1 = blk % CM1;
  const float th = conf_th[c1];
  const float* src = probs + (size_t)blk * N;
  const int nTiles = (N + TILE - 1) / TILE;
  const uint32_t row_base = lds_off32(row);

  auto issue = [&](int t) {
    int e0 = t * TILE + (int)threadIdx.x * 4;        // lane's element in row
    int ec = e0;                                     // clamp tail global addr
    if (ec > N - 4) ec = (N >= 4) ? (N - 4) : 0;
    // LDS slot keeps the true lane position (pad region absorbs tail dupes)
    async_load_b128(row_base + (uint32_t)e0 * 4u, src + ec);
  };

  // ---- clear hist, then DMA whole row with pass-1 histogram fused ------------
  for (int i = threadIdx.x; i < NBIN; i += TPB) hist[i] = 0u;
  __syncthreads();

  {
    int pre = nTiles < DEPTH ? nTiles : DEPTH;
    for (int t = 0; t < pre; ++t) issue(t);
    for (int t = 0; t < nTiles; ++t) {
      if (t + DEPTH < nTiles) issue(t + DEPTH);
      int rem = nTiles - 1 - t; if (rem > DEPTH) rem = DEPTH;
      wait_async(rem);          // tile t complete (in-order completion)
      __syncthreads();          // all waves' pieces of tile t visible
      int base = t * TILE;
      int lim = N - base; if (lim > TILE) lim = TILE;
      for (int i = (int)threadIdx.x; i < lim; i += TPB) {
        float x = row[base + i];
        if (x > th) atomicAdd(&hist[__float_as_uint(x) >> 19], 1u);
      }
    }
  }
  __syncthreads();

  // ---- select crossing bin of bits [30:19] -----------------------------------
  if (threadIdx.x == 0) {
    uint32_t cum = 0; int bsel = -1;
    for (int bi = NBIN - 1; bi >= 1; --bi) {
      uint32_t h = hist[bi];
      if (cum + h >= (uint32_t)K) { bsel = bi; break; }
      cum += h;
    }
    if (bsel < 0) { s_thresh = 1u; s_flag = 0u; s_b1 = 0u; s_cAbove = cum; }
    else          { s_b1 = (uint32_t)bsel; s_cAbove = cum; s_flag = 1u; }
  }
  __syncthreads();
  const uint32_t b1 = s_b1;

  // ---- pass 2 (LDS only): refine bits [18:7] inside crossing bin -------------
  if (s_flag) {
    for (int i = threadIdx.x; i < NBIN; i += TPB) hist[i] = 0u;
    __syncthreads();
    for (int i = (int)threadIdx.x; i < N; i += TPB) {
      float x = row[i];
      if (x > th) {
        uint32_t u = __float_as_uint(x);
        if ((u >> 19) == b1) atomicAdd(&hist[(u >> 7) & 0xFFFu], 1u);
      }
    }
    __syncthreads();
    if (threadIdx.x == 0) {
      uint32_t cum = s_cAbove; uint32_t ssel = 0;
      for (int si = NBIN - 1; si >= 0; --si) {
        uint32_t h = hist[si];
        if (cum + h >= (uint32_t)K) { ssel = (uint32_t)si; break; }
        cum += h;
      }
      s_thresh = (b1 << 19) | (ssel << 7);
    }
    __syncthreads();
  }
  const uint32_t thresh = s_thresh;

  // ---- pass 3 (LDS only): compact candidates >= 24-bit threshold -------------
  if (threadIdx.x == 0) s_cnt = 0u;
  __syncthreads();
  for (int i = (int)threadIdx.x; i < N; i += TPB) {
    float x = row[i];
    if (x > th) {
      uint32_t u = __float_as_uint(x);
      if (u >= thresh) {
        uint32_t pos = atomicAdd(&s_cnt, 1u);
        if (pos < CANDN)
          cand[pos] = ((unsigned long long)(~u) << 32) | (uint32_t)i;
      }
    }
  }
  __syncthreads();
  uint32_t cnt = s_cnt; if (cnt > CANDN) cnt = CANDN;
  for (int i = threadIdx.x; i < CANDN; i += TPB)
    if ((uint32_t)i >= cnt)
      cand[i] = (0xFFFFFFFFull << 32) | (uint32_t)i;  // key=0 sentinel, sorts last
  __syncthreads();

  // ---- bitonic sort ascending on composite (~key, idx) -----------------------
  for (unsigned k = 2; k <= CANDN; k <<= 1) {
    for (unsigned j = k >> 1; j > 0; j >>= 1) {
      for (unsigned i = threadIdx.x; i < CANDN; i += TPB) {
        unsigned ixj = i ^ j;
        if (ixj > i) {
          bool dir = ((i & k) == 0);
          unsigned long long a = cand[i], b = cand[ixj];
          if ((a > b) == dir) { cand[i] = b; cand[ixj] = a; }
        }
      }
      __syncthreads();
    }
  }

  // ---- emit top-K -------------------------------------------------------------
  for (int kk = threadIdx.x; kk < K; kk += TPB) {
    unsigned long long q = cand[kk];
    uint32_t key = ~(uint32_t)(q >> 32);
    bool valid = (kk < (int)cnt) && key != 0u;
    tk_sc[(size_t)blk * K + kk] = valid ? __uint_as_float(key) : 0.f;
    tk_id[(size_t)blk * K + kk] = valid ? (int)(uint32_t)q : -1;
  }
}

// ------------------------------ kernel 3 -----------------------------------
__global__ void __launch_bounds__(TPB)
nms_kernel(const float* __restrict__ boxes,   // [B,N,4]
           const float* __restrict__ tk_sc,   // [B*CM1, K]
           const int* __restrict__ tk_id,     // [B*CM1, K]
           const float* __restrict__ nms_th_p,
           float* __restrict__ out,           // boxes|labels|scores|keep concat
           int B, int CM1, int N, int K) {
  __shared__ float Ls[KMAX], Ts[KMAX], Rs[KMAX], Bs[KMAX], As[KMAX], Ss[KMAX];
  __shared__ int alive[KMAX];
  const int blk = blockIdx.x;
  const int b = blk / CM1, c1 = blk % CM1;
  const int j = threadIdx.x;

  if (j < K) {
    int id = tk_id[(size_t)blk * K + j];
    float sc = tk_sc[(size_t)blk * K + j];
    float l = 0.f, t = 0.f, r = 0.f, bo = 0.f;
    if (id >= 0) {
      const float* bp = boxes + ((size_t)b * N + (size_t)id) * 4;
      l = bp[0]; t = bp[1]; r = bp[2]; bo = bp[3];
    }
    Ls[j] = l; Ts[j] = t; Rs[j] = r; Bs[j] = bo;
    As[j] = fmaxf(r - l, 0.f) * fmaxf(bo - t, 0.f);
    Ss[j] = sc;
    alive[j] = (id >= 0) ? 1 : 0;
  }
  __syncthreads();

  const float nth = *nms_th_p;
  for (int i = 0; i < K; ++i) {
    bool keep_i = (alive[i] != 0);
    if (keep_i && j > i && j < K && alive[j]) {
      float lw = fmaxf(fminf(Rs[i], Rs[j]) - fmaxf(Ls[i], Ls[j]), 0.f);
      float lh = fmaxf(fminf(Bs[i], Bs[j]) - fmaxf(Ts[i], Ts[j]), 0.f);
      float inter = lw * lh;
      float iou = inter / (As[i] + As[j] - inter + 1e-9f);
      if (iou >= nth) alive[j] = 0;
    }
    __syncthreads();
  }

  const int M = CM1 * K;
  if (j < K) {
    bool kp = (alive[j] != 0);
    size_t m = (size_t)b * M + (size_t)c1 * K + j;
    float* ob = out;
    float* ol = out + (size_t)B * M * 4;
    float* os = ol + (size_t)B * M;
    float* ok = os + (size_t)B * M;
    ob[m * 4 + 0] = kp ? Ls[j] : 0.f;
    ob[m * 4 + 1] = kp ? Ts[j] : 0.f;
    ob[m * 4 + 2] = kp ? Rs[j] : 0.f;
    ob[m * 4 + 3] = kp ? Bs[j] : 0.f;
    ol[m] = kp ? (float)(c1 + 1) : 0.f;
    os[m] = kp ? Ss[j] : 0.f;
    ok[m] = kp ? 1.f : 0.f;
  }
}

// ------------------------------ launcher -----------------------------------
extern "C" void kernel_launch(void* const* d_in, const int* in_sizes, int n_in,
                              void* d_out, int out_size, void* d_ws, size_t ws_size,
                              hipStream_t stream) {
  const float* bb_in = (const float*)d_in[0];  // [B,4,N]
  const float* sc_in = (const float*)d_in[1];  // [B,C,N]
  const float* dbox  = (const float*)d_in[2];  // [1,N,4]
  const float* cth   = (const float*)d_in[3];  // [C-1]
  const float* nth   = (const float*)d_in[4];  // [1]
  // d_in[5] = max_num (int, device); K derived from out_size instead
  const float* sxy   = (const float*)d_in[6];
  const float* swh   = (const float*)d_in[7];
  (void)n_in; (void)ws_size;

  const int CM1 = in_sizes[3];
  const int C   = CM1 + 1;
  const int N   = in_sizes[2] / 4;
  const int B   = in_sizes[1] / (C * N);
  int K = out_size / (7 * B * CM1);
  if (K > KMAX) K = KMAX;
  if (K < 1) K = 1;

  float* boxes = (float*)d_ws;                       // B*N*4
  float* probs = boxes + (size_t)B * N * 4;          // B*CM1*N
  float* tk_sc = probs + (size_t)B * CM1 * N;        // B*CM1*K
  int*   tk_id = (int*)(tk_sc + (size_t)B * CM1 * K);

  const int total = B * N;
  dim3 g1((total + TPB - 1) / TPB), blkd(TPB);
  if (C == 21) {
    decode_softmax_kernel<21><<<g1, blkd, 0, stream>>>(bb_in, sc_in, dbox, sxy, swh,
                                                       boxes, probs, B, N);
  } else {
    decode_softmax_dyn<<<g1, blkd, 0, stream>>>(bb_in, sc_in, dbox, sxy, swh,
                                                boxes, probs, B, N, C);
  }

  const int nTiles = (N + TILE - 1) / TILE;
  // dynamic LDS: hist (16KB) + cand (8KB) + padded row (~96KB) ≈ 123KB
  // (MI455X WGP allows up to 320KB per workgroup)
  const size_t smem2 = (size_t)NBIN * 4 + (size_t)CANDN * 8 +
                       (size_t)nTiles * TILE * 4;
  topk_kernel<<<B * CM1, TPB, smem2, stream>>>(probs, cth, tk_sc, tk_id, CM1, N, K);

  nms_kernel<<<B * CM1, TPB, 0, stream>>>(boxes, tk_sc, tk_id, nth,
                                          (float*)d_out, B, CM1, N, K);
}